// CamAttnCon_16484084483308
// MI455X (gfx1250) — compile-verified
//
#include <hip/hip_runtime.h>
#include <hip/hip_bf16.h>
#include <math.h>

typedef __attribute__((ext_vector_type(2))) float v2f;
typedef __attribute__((ext_vector_type(8))) float v8f;

// Problem constants (from reference)
#define BB 32
#define HH 8
#define TT 512
#define SS 196
#define DD 768
#define KK 51   // int(0.1 * 512)

// ---------------------------------------------------------------------------
// Kernel 1: cosine weights via V_WMMA_F32_16X16X4_F32.
// One wave (32 threads) per 16-row tile of target_embed. grid = (T/16, B).
//   A (16x4 f32): lane l holds row m=l%16, K = {0,1}+2*(l>=16)  -> v2f load
//   B (4x16 f32): all N columns = x chunk -> D[m][n] == dot(row m) for all n
// Row sum-of-squares and ||x||^2 accumulate in VALU alongside.
// ---------------------------------------------------------------------------
__global__ __launch_bounds__(32) void cam_weights_kernel(
    const float* __restrict__ target,   // (B,T,D)
    const float* __restrict__ fore,     // (B,D)
    float* __restrict__ w_out)          // (B,T)
{
    const int tile = blockIdx.x;        // 0..T/16-1
    const int b    = blockIdx.y;        // 0..B-1
    const int lane = threadIdx.x;       // 0..31
    const int m    = lane & 15;
    const int half = lane >> 4;         // 0 or 1

    const float* Arow = target + ((size_t)b * TT + (size_t)tile * 16 + m) * DD + 2 * half;
    const float* X    = fore   + (size_t)b * DD + 2 * half;

    v8f  c   = {};
    float nsq = 0.0f;   // this lane's half of row m's sum of squares
    float ysq = 0.0f;   // this lane's half of ||x||^2

    #pragma unroll 4
    for (int k = 0; k < DD; k += 4) {
        v2f a = *(const v2f*)(Arow + k);   // A: K = k + {0,1} + 2*half
        v2f x = *(const v2f*)(X    + k);   // B: same K mapping, all columns
        c = __builtin_amdgcn_wmma_f32_16x16x4_f32(
                /*neg_a=*/false, a, /*neg_b=*/false, x,
                /*c_mod=*/(short)0, c, /*reuse_a=*/false, /*reuse_b=*/false);
        nsq = fmaf(a.x, a.x, fmaf(a.y, a.y, nsq));
        ysq = fmaf(x.x, x.x, fmaf(x.y, x.y, ysq));
    }

    // Combine the two K-halves: partner lane is l ^ 16.
    nsq += __shfl_xor(nsq, 16, 32);   // lane l now has full sumsq of row l%16
    ysq += __shfl_xor(ysq, 16, 32);   // every lane: full ||x||^2

    __shared__ float s_dot[16];
    __shared__ float s_nsq[16];

    if (half == 0) s_nsq[m] = nsq;
    // D layout: lane with N=0 holds rows (v + 8*half) in c[v]; cols identical.
    if (m == 0) {
        s_dot[8 * half + 0] = c[0];
        s_dot[8 * half + 1] = c[1];
        s_dot[8 * half + 2] = c[2];
        s_dot[8 * half + 3] = c[3];
        s_dot[8 * half + 4] = c[4];
        s_dot[8 * half + 5] = c[5];
        s_dot[8 * half + 6] = c[6];
        s_dot[8 * half + 7] = c[7];
    }
    __syncthreads();

    if (half == 0) {
        float dot = s_dot[m];
        float nx  = sqrtf(s_nsq[m]);
        float ny  = sqrtf(ysq);
        float w   = dot / fmaxf(nx * ny, 1e-8f);
        w_out[(size_t)b * TT + (size_t)tile * 16 + m] = w;
    }
}

// ---------------------------------------------------------------------------
// Kernel 2: per-batch top-51 of 512 (descending, ties -> lower index, matching
// jax.lax.top_k). One 64-thread block per batch; iterative argmax selection.
// ---------------------------------------------------------------------------
__global__ __launch_bounds__(64) void cam_topk_kernel(
    const float* __restrict__ w,        // (B,T)
    int* __restrict__ idx_out,          // (B,K) for kernel 3
    float* __restrict__ idx_f_out)      // (B,K) float copy into d_out
{
    const int b   = blockIdx.x;
    const int tid = threadIdx.x;        // 0..63

    __shared__ float vals[TT];
    __shared__ float rv[64];
    __shared__ int   ri[64];

    for (int i = tid; i < TT; i += 64) vals[i] = w[(size_t)b * TT + i];
    __syncthreads();

    for (int j = 0; j < KK; ++j) {
        float best = -INFINITY;
        int   bi   = TT;
        for (int i = tid; i < TT; i += 64) {
            float v = vals[i];
            if (v > best) { best = v; bi = i; }   // ascending i: keeps min idx on tie
        }
        rv[tid] = best;
        ri[tid] = bi;
        __syncthreads();
        for (int s = 32; s > 0; s >>= 1) {
            if (tid < s) {
                float v2 = rv[tid + s];
                int   i2 = ri[tid + s];
                if (v2 > rv[tid] || (v2 == rv[tid] && i2 < ri[tid])) {
                    rv[tid] = v2; ri[tid] = i2;
                }
            }
            __syncthreads();
        }
        if (tid == 0) {
            int sel = ri[0];
            idx_out[b * KK + j]   = sel;
            idx_f_out[b * KK + j] = (float)sel;
            vals[sel] = -INFINITY;
        }
        __syncthreads();
    }
}

// ---------------------------------------------------------------------------
// Kernel 3: for the 51 selected rows only: mean over 8 heads, scale by weight,
// ReLU, per-row (x - min)/clip(max(x-min),1e-12,1), running max over rows.
// Also copies fore_map into the first output segment.
// One 256-thread block per batch (thread s covers column s < 196).
// Reads only 51/512 of the attention rows: ~10 MB instead of ~103 MB.
// ---------------------------------------------------------------------------
__global__ __launch_bounds__(256) void cam_gather_norm_kernel(
    const float* __restrict__ attn,     // (3,B,H,T,S) -> layer 2
    const float* __restrict__ fore_map, // (B,1,S)
    const float* __restrict__ w,        // (B,T)
    const int* __restrict__ idxs,       // (B,K)
    float* __restrict__ out)            // [fm | total_attn | idx_f]
{
    const int b   = blockIdx.x;
    const int tid = threadIdx.x;

    __shared__ float red[256];

    const size_t layer_base =
        ((size_t)2 * BB + (size_t)b) * (size_t)HH * TT * SS;

    float running = -INFINITY;

    for (int j = 0; j < KK; ++j) {
        const int   t  = idxs[b * KK + j];
        const float wj = w[(size_t)b * TT + t];

        float v = 0.0f;
        if (tid < SS) {
            float acc = 0.0f;
            const size_t base = layer_base + (size_t)t * SS + tid;
            #pragma unroll
            for (int h = 0; h < HH; ++h)
                acc += attn[base + (size_t)h * TT * SS];
            v = fmaxf(wj * (acc * (1.0f / HH)), 0.0f);   // relu(w * mean_h)
        }

        // min over s
        red[tid] = (tid < SS) ? v : INFINITY;
        __syncthreads();
        for (int s = 128; s > 0; s >>= 1) {
            if (tid < s) red[tid] = fminf(red[tid], red[tid + s]);
            __syncthreads();
        }
        const float mn = red[0];
        __syncthreads();

        const float vm = v - mn;

        // max over s of (v - min)
        red[tid] = (tid < SS) ? vm : -INFINITY;
        __syncthreads();
        for (int s = 128; s > 0; s >>= 1) {
            if (tid < s) red[tid] = fmaxf(red[tid], red[tid + s]);
            __syncthreads();
        }
        const float mx = red[0];
        __syncthreads();

        const float amax = fminf(fmaxf(mx, 1e-12f), 1.0f);
        running = fmaxf(running, vm / amax);
    }

    if (tid < SS) {
        out[(size_t)BB * SS + (size_t)b * SS + tid] = running;          // total_attn
        out[(size_t)b * SS + tid] = fore_map[(size_t)b * SS + tid];     // fm
    }
}

// ---------------------------------------------------------------------------
// Launch: weights (WMMA) -> top-k -> gather/normalize/output, stream-ordered.
// ---------------------------------------------------------------------------
extern "C" void kernel_launch(void* const* d_in, const int* in_sizes, int n_in,
                              void* d_out, int out_size, void* d_ws, size_t ws_size,
                              hipStream_t stream) {
    (void)in_sizes; (void)n_in; (void)out_size; (void)ws_size;

    const float* fore_map = (const float*)d_in[0];   // (B,1,S)
    const float* fore_rep = (const float*)d_in[1];   // (B,D)
    const float* target   = (const float*)d_in[2];   // (B,T,D)
    const float* attns    = (const float*)d_in[3];   // (3,B,H,T,S)

    float* out = (float*)d_out;
    // d_out layout: [0, B*S) fm | [B*S, 2*B*S) total_attn | [2*B*S, +B*K) idxs
    float* out_idx_f = out + (size_t)2 * BB * SS;

    float* ws_w   = (float*)d_ws;                                   // B*T floats
    int*   ws_idx = (int*)((char*)d_ws + (size_t)BB * TT * sizeof(float)); // B*K ints

    dim3 g1(TT / 16, BB);
    cam_weights_kernel<<<g1, 32, 0, stream>>>(target, fore_rep, ws_w);
    cam_topk_kernel<<<BB, 64, 0, stream>>>(ws_w, ws_idx, out_idx_f);
    cam_gather_norm_kernel<<<BB, 256, 0, stream>>>(attns, fore_map, ws_w, ws_idx, out);
}